// SRNNPredictor_89103391522909
// MI455X (gfx1250) — compile-verified
//
#include <hip/hip_runtime.h>
#include <math.h>

typedef float v2f __attribute__((ext_vector_type(2)));
typedef float v8f __attribute__((ext_vector_type(8)));

#define VOC   50257
#define INP   512
#define HID   1024
#define NSTACK 4
#define DEPTH  8
#define SSIZE  64
#define EMPTY_VAL (-1.0f)

// -------------------------------------------------------------------------
// K1: mid[i] = tanh( emb·W_i2h[i] + h·W_h2h[i] + biases + stack terms )
// One wave (32 lanes) per output row; 1024 waves total.
// -------------------------------------------------------------------------
__global__ void __launch_bounds__(256)
k_mid(const int* __restrict__ token,
      const float* __restrict__ hidden,
      const float* __restrict__ stacks,
      const float* __restrict__ emb_W,
      const float* __restrict__ W_i2h, const float* __restrict__ b_i2h,
      const float* __restrict__ W_h2h, const float* __restrict__ b_h2h,
      const float* __restrict__ W_s2h, const float* __restrict__ b_s2h,
      float* __restrict__ mid_ws,       // ws: tanh(mid), read by K3
      float* __restrict__ hid_out)      // d_out + VOC
{
    const int wave = (blockIdx.x * blockDim.x + threadIdx.x) >> 5; // 0..1023
    const int lane = threadIdx.x & 31;
    const int i = wave;
    if (i >= HID) return;

    const int t = token[0];
    const float* __restrict__ emb = emb_W + (size_t)t * INP;
    const float* __restrict__ wi  = W_i2h + (size_t)i * INP;
    const float* __restrict__ wh  = W_h2h + (size_t)i * HID;

    float acc = 0.0f;
    #pragma unroll 4
    for (int j = lane; j < INP; j += 32) acc += wi[j] * emb[j];
    #pragma unroll 4
    for (int j = lane; j < HID; j += 32) acc += wh[j] * hidden[j];

    #pragma unroll
    for (int off = 16; off > 0; off >>= 1) acc += __shfl_down(acc, off, 32);

    if (lane == 0) {
        float s = b_i2h[i] + b_h2h[i];
        #pragma unroll
        for (int n = 0; n < NSTACK; ++n) {
            s += b_s2h[n * HID + i];
            const float* __restrict__ ws = W_s2h + ((size_t)n * HID + i) * DEPTH;
            const float* __restrict__ st = stacks + n * SSIZE;
            #pragma unroll
            for (int d = 0; d < DEPTH; ++d) s += ws[d] * st[d];
        }
        const float m = tanhf(acc + s);
        mid_ws[i]  = m;
        hid_out[i] = m;
    }
}

// -------------------------------------------------------------------------
// K2: action softmax + stack update (push-value softmax over size-1 == 1.0)
// Single block, 384 threads (12 waves, one per (stack, action) logit).
// -------------------------------------------------------------------------
__global__ void __launch_bounds__(384)
k_stacks(const float* __restrict__ hidden,
         const float* __restrict__ stacks,
         const float* __restrict__ W_act, const float* __restrict__ b_act,
         float* __restrict__ stacks_out)   // d_out + VOC + HID
{
    __shared__ float s_logit[NSTACK * 3];
    __shared__ float s_act[NSTACK * 3];

    const int tid  = threadIdx.x;
    const int wave = tid >> 5;
    const int lane = tid & 31;

    if (wave < NSTACK * 3) {
        const float* __restrict__ w = W_act + (size_t)wave * HID;
        float acc = 0.0f;
        #pragma unroll 4
        for (int j = lane; j < HID; j += 32) acc += w[j] * hidden[j];
        #pragma unroll
        for (int off = 16; off > 0; off >>= 1) acc += __shfl_down(acc, off, 32);
        if (lane == 0) s_logit[wave] = acc + b_act[wave];
    }
    __syncthreads();

    if (tid < NSTACK) {
        const float l0 = s_logit[tid * 3 + 0];
        const float l1 = s_logit[tid * 3 + 1];
        const float l2 = s_logit[tid * 3 + 2];
        const float mx = fmaxf(l0, fmaxf(l1, l2));
        const float e0 = expf(l0 - mx), e1 = expf(l1 - mx), e2 = expf(l2 - mx);
        const float inv = 1.0f / (e0 + e1 + e2);
        s_act[tid * 3 + 0] = e0 * inv;
        s_act[tid * 3 + 1] = e1 * inv;
        s_act[tid * 3 + 2] = e2 * inv;
    }
    __syncthreads();

    if (tid < NSTACK * SSIZE) {
        const int n = tid >> 6;       // /SSIZE
        const int s = tid & 63;       // %SSIZE
        const float a0 = s_act[n * 3 + 0];
        const float a1 = s_act[n * 3 + 1];
        const float a2 = s_act[n * 3 + 2];
        const float* __restrict__ st = stacks + n * SSIZE;
        float v;
        if (s == 0)                 v = a0 * 1.0f;          // push value: softmax over size-1 dim == 1
        else if (s == SSIZE - 1)    v = a1 * EMPTY_VAL;
        else                        v = a0 * st[s - 1] + a1 * st[s + 1] + a2 * st[s];
        stacks_out[n * SSIZE + s] = v;
    }
}

// -------------------------------------------------------------------------
// K3: logits = W_h2o @ mid + b  via V_WMMA_F32_16X16X4_F32.
// One wave per 16-row tile of W_h2o. A = 16x4 weight tile (doc layout:
// lanes 0-15 carry K=0,1; lanes 16-31 carry K=2,3). B = mid chunk broadcast
// across all N columns -> every column of C holds the same dot product.
// Lanes 0/16 hold M=0..7 / M=8..15 of column 0 per the C/D layout.
// EXEC stays all-1s: tail tiles clamp their row index and skip the store.
// -------------------------------------------------------------------------
#define ROW_TILES ((VOC + 15) / 16)   // 3142

__global__ void __launch_bounds__(256)
k_logits_wmma(const float* __restrict__ W_h2o,
              const float* __restrict__ b_h2o,
              const float* __restrict__ mid_ws,
              float* __restrict__ logits_ws)
{
    const int wave = (blockIdx.x * blockDim.x + threadIdx.x) >> 5;
    const int lane = threadIdx.x & 31;

    const int row_base = wave * 16;
    const int m        = lane & 15;
    const int khalf    = (lane >> 4) << 1;   // 0 for lanes 0-15, 2 for lanes 16-31

    int row = row_base + m;
    if (row > VOC - 1) row = VOC - 1;        // clamp: no OOB loads, EXEC stays full
    const float* __restrict__ wrow = W_h2o + (size_t)row * HID;

    v8f c = {};
    #pragma unroll 8
    for (int k0 = 0; k0 < HID; k0 += 4) {
        const v2f a = *(const v2f*)(wrow   + k0 + khalf);   // A: M=lane&15, K=khalf..khalf+1
        const v2f b = *(const v2f*)(mid_ws + k0 + khalf);   // B: row K broadcast across N
        c = __builtin_amdgcn_wmma_f32_16x16x4_f32(
                /*neg_a=*/false, a, /*neg_b=*/false, b,
                /*c_mod=*/(short)0, c, /*reuse_a=*/false, /*reuse_b=*/false);
    }

    if ((lane & 15) == 0) {                   // lane 0 -> M=0..7, lane 16 -> M=8..15
        const int mbase = row_base + ((lane >> 4) << 3);
        #pragma unroll
        for (int j = 0; j < 8; ++j) {
            const int r = mbase + j;
            if (r < VOC) logits_ws[r] = c[j] + b_h2o[r];
        }
    }
}

// -------------------------------------------------------------------------
// K4: log_softmax over 50257 logits, single resident block of 1024 threads.
// Three strided passes over L2-resident logits (~200 KB).
// -------------------------------------------------------------------------
__global__ void __launch_bounds__(1024)
k_logsoftmax(const float* __restrict__ logits_ws, float* __restrict__ out)
{
    __shared__ float red[32];
    const int tid = threadIdx.x;

    // pass 1: max
    float mx = -INFINITY;
    for (int i = tid; i < VOC; i += 1024) mx = fmaxf(mx, logits_ws[i]);
    #pragma unroll
    for (int off = 16; off > 0; off >>= 1) mx = fmaxf(mx, __shfl_down(mx, off, 32));
    if ((tid & 31) == 0) red[tid >> 5] = mx;
    __syncthreads();
    if (tid < 32) {
        float v = red[tid];
        #pragma unroll
        for (int off = 16; off > 0; off >>= 1) v = fmaxf(v, __shfl_down(v, off, 32));
        if (tid == 0) red[0] = v;
    }
    __syncthreads();
    mx = red[0];
    __syncthreads();

    // pass 2: sum of exp
    float s = 0.0f;
    for (int i = tid; i < VOC; i += 1024) s += expf(logits_ws[i] - mx);
    #pragma unroll
    for (int off = 16; off > 0; off >>= 1) s += __shfl_down(s, off, 32);
    if ((tid & 31) == 0) red[tid >> 5] = s;
    __syncthreads();
    if (tid < 32) {
        float v = red[tid];
        #pragma unroll
        for (int off = 16; off > 0; off >>= 1) v += __shfl_down(v, off, 32);
        if (tid == 0) red[0] = v;
    }
    __syncthreads();
    const float lse = mx + logf(red[0]);

    // pass 3: write
    for (int i = tid; i < VOC; i += 1024) out[i] = logits_ws[i] - lse;
}

// -------------------------------------------------------------------------
// Launch
// -------------------------------------------------------------------------
extern "C" void kernel_launch(void* const* d_in, const int* in_sizes, int n_in,
                              void* d_out, int out_size, void* d_ws, size_t ws_size,
                              hipStream_t stream)
{
    (void)in_sizes; (void)n_in; (void)out_size; (void)ws_size;

    const int*   token  = (const int*)  d_in[0];
    const float* hidden = (const float*)d_in[1];
    const float* stacks = (const float*)d_in[2];
    const float* emb_W  = (const float*)d_in[3];
    const float* W_i2h  = (const float*)d_in[4];
    const float* b_i2h  = (const float*)d_in[5];
    const float* W_h2h  = (const float*)d_in[6];
    const float* b_h2h  = (const float*)d_in[7];
    const float* W_s2h  = (const float*)d_in[8];
    const float* b_s2h  = (const float*)d_in[9];
    const float* W_act  = (const float*)d_in[10];
    const float* b_act  = (const float*)d_in[11];
    // d_in[12] W_push, d_in[13] b_push: softmax over a size-1 dim is identically 1.0 -> unused
    const float* W_h2o  = (const float*)d_in[14];
    const float* b_h2o  = (const float*)d_in[15];

    float* out_logsm  = (float*)d_out;                 // [VOC]
    float* out_hidden = (float*)d_out + VOC;           // [HID]
    float* out_stacks = (float*)d_out + VOC + HID;     // [NSTACK*SSIZE]

    float* mid_ws    = (float*)d_ws;                           // 1024 floats
    float* logits_ws = (float*)((char*)d_ws + 4096);           // 50257 floats

    // K1: mid + tanh (1024 waves = 128 blocks x 256 threads)
    k_mid<<<dim3(128), dim3(256), 0, stream>>>(
        token, hidden, stacks, emb_W, W_i2h, b_i2h, W_h2h, b_h2h,
        W_s2h, b_s2h, mid_ws, out_hidden);

    // K2: action softmax + stack update (independent of K1; uses OLD hidden)
    k_stacks<<<dim3(1), dim3(384), 0, stream>>>(
        hidden, stacks, W_act, b_act, out_stacks);

    // K3: WMMA GEMV, 3142 tiles, 8 waves per block -> 393 blocks
    k_logits_wmma<<<dim3((ROW_TILES + 7) / 8), dim3(256), 0, stream>>>(
        W_h2o, b_h2o, mid_ws, logits_ws);

    // K4: log_softmax
    k_logsoftmax<<<dim3(1), dim3(1024), 0, stream>>>(logits_ws, out_logsm);
}